// TransO_20151986552886
// MI455X (gfx1250) — compile-verified
//
#include <hip/hip_runtime.h>
#include <math.h>

#define D      128
#define BATCH  16384
#define NET    100          // entity types
#define NRT    50           // relation types
#define COLS   64           // items per block (4 column tiles of 16)
#define CT     4            // column tiles
#define MAXT   ((BATCH + COLS - 1) / COLS)   // 256 tiles worst case per type

typedef float v2f __attribute__((ext_vector_type(2)));
typedef float v8f __attribute__((ext_vector_type(8)));

// ---------------- bucketing kernels ----------------

__global__ void k_init(int* cntE, int* cntR) {
    int t = threadIdx.x;
    if (t < NET) cntE[t] = 0;
    if (t < NRT) cntR[t] = 0;
}

__global__ void k_count(const int* __restrict__ et, const int* __restrict__ rt,
                        int* cntE, int* cntR) {
    int b = blockIdx.x * blockDim.x + threadIdx.x;
    if (b < BATCH) {
        atomicAdd(&cntE[et[b]], 1);
        atomicAdd(&cntR[rt[b]], 1);
    }
}

__global__ void k_scan(const int* cntE, int* offE, int* curE,
                       const int* cntR, int* offR, int* curR) {
    if (threadIdx.x == 0) {
        int s = 0;
        for (int i = 0; i < NET; ++i) { offE[i] = s; curE[i] = s; s += cntE[i]; }
    }
    if (threadIdx.x == 1) {
        int s = 0;
        for (int i = 0; i < NRT; ++i) { offR[i] = s; curR[i] = s; s += cntR[i]; }
    }
}

__global__ void k_scatter(const int* __restrict__ et, const int* __restrict__ rt,
                          int* curE, int* curR, int* permE, int* permR) {
    int b = blockIdx.x * blockDim.x + threadIdx.x;
    if (b < BATCH) {
        permE[atomicAdd(&curE[et[b]], 1)] = b;
        permR[atomicAdd(&curR[rt[b]], 1)] = b;
    }
}

// ---------------- WMMA GEMM pass 1: U[b] = Pe[type] @ (h[b] - t[b]) ----------------

__global__ __launch_bounds__(128)
void k_gemm_e(const int* __restrict__ heads, const int* __restrict__ tails,
              const float* __restrict__ eemb, const float* __restrict__ eproj,
              const int* __restrict__ cntE, const int* __restrict__ offE,
              const int* __restrict__ permE, float* __restrict__ U) {
    const int type  = blockIdx.x;
    const int tileI = blockIdx.y;
    const int cnt   = cntE[type];
    if (tileI * COLS >= cnt) return;
    const int off = offE[type];

    __shared__ float X[CT][D][16];   // [col-tile][k][col within tile]
    __shared__ int   bcol[COLS];
    __shared__ int   vcol[COLS];

    const int tid   = threadIdx.x;
    const int wave  = tid >> 5;
    const int lane  = tid & 31;
    const int lhalf = lane >> 4;   // selects K pair within K=4 step
    const int l15   = lane & 15;
    const float* P  = eproj + (size_t)type * (D * D);

    // warm WGP$ with this lane's two A rows while we stage X
    const float* arow0 = P + (size_t)(wave * 16 + l15) * D;
    const float* arow1 = P + (size_t)((wave + 4) * 16 + l15) * D;
    __builtin_prefetch(arow0, 0, 3);
    __builtin_prefetch(arow1, 0, 3);

    {   // stage X = h - t for 64 items
        const int c = tid & 63;        // column (item)
        const int q = tid >> 6;        // 0..1, each loads 64 consecutive k
        const int item  = tileI * COLS + c;
        const int valid = item < cnt;
        const int b = valid ? permE[off + item] : 0;
        if (q == 0) { bcol[c] = b; vcol[c] = valid; }
        const float* hp = eemb + (size_t)heads[b] * D;
        const float* tp = eemb + (size_t)tails[b] * D;
        const int cc = c >> 4, c16 = c & 15;
        #pragma unroll 16
        for (int jj = 0; jj < 64; ++jj) {
            const int j = q * 64 + jj;
            X[cc][j][c16] = valid ? (hp[j] - tp[j]) : 0.0f;
        }
    }
    __syncthreads();

    v8f acc[2][CT];
    #pragma unroll
    for (int rr = 0; rr < 2; ++rr)
        #pragma unroll
        for (int cc = 0; cc < CT; ++cc)
            acc[rr][cc] = (v8f){0.f,0.f,0.f,0.f,0.f,0.f,0.f,0.f};

    #pragma unroll 4
    for (int k0 = 0; k0 < D; k0 += 4) {
        const int kb = k0 + lhalf * 2;
        v2f a0, a1;
        a0[0] = arow0[kb]; a0[1] = arow0[kb + 1];
        a1[0] = arow1[kb]; a1[1] = arow1[kb + 1];
        #pragma unroll
        for (int cc = 0; cc < CT; ++cc) {
            v2f bb;
            bb[0] = X[cc][kb][l15];
            bb[1] = X[cc][kb + 1][l15];
            acc[0][cc] = __builtin_amdgcn_wmma_f32_16x16x4_f32(
                false, a0, false, bb, (short)0, acc[0][cc], false, false);
            acc[1][cc] = __builtin_amdgcn_wmma_f32_16x16x4_f32(
                false, a1, false, bb, (short)0, acc[1][cc], false, false);
        }
    }

    #pragma unroll
    for (int rr = 0; rr < 2; ++rr) {
        const int rt = wave + rr * 4;
        #pragma unroll
        for (int cc = 0; cc < CT; ++cc) {
            const int c = cc * 16 + l15;
            if (vcol[c]) {
                float* ub = U + (size_t)bcol[c] * D + rt * 16 + lhalf * 8;
                #pragma unroll
                for (int g = 0; g < 8; ++g) ub[g] = acc[rr][cc][g];
            }
        }
    }
}

// -------- WMMA GEMM pass 2: out[b] = || U[b] + Pr[type] @ r[b] || --------

__global__ __launch_bounds__(128)
void k_gemm_r(const int* __restrict__ rels,
              const float* __restrict__ remb, const float* __restrict__ rproj,
              const int* __restrict__ cntR, const int* __restrict__ offR,
              const int* __restrict__ permR, const float* __restrict__ U,
              float* __restrict__ out) {
    const int type  = blockIdx.x;
    const int tileI = blockIdx.y;
    const int cnt   = cntR[type];
    if (tileI * COLS >= cnt) return;
    const int off = offR[type];

    __shared__ float X[CT][D][16];
    __shared__ int   bcol[COLS];
    __shared__ int   vcol[COLS];
    __shared__ float psum[8][COLS];   // [wave*2+lhalf][col] deterministic reduce

    const int tid   = threadIdx.x;
    const int wave  = tid >> 5;
    const int lane  = tid & 31;
    const int lhalf = lane >> 4;
    const int l15   = lane & 15;
    const float* P  = rproj + (size_t)type * (D * D);

    const float* arow0 = P + (size_t)(wave * 16 + l15) * D;
    const float* arow1 = P + (size_t)((wave + 4) * 16 + l15) * D;
    __builtin_prefetch(arow0, 0, 3);
    __builtin_prefetch(arow1, 0, 3);

    {
        const int c = tid & 63;
        const int q = tid >> 6;
        const int item  = tileI * COLS + c;
        const int valid = item < cnt;
        const int b = valid ? permR[off + item] : 0;
        if (q == 0) { bcol[c] = b; vcol[c] = valid; }
        const float* rp = remb + (size_t)rels[b] * D;
        const int cc = c >> 4, c16 = c & 15;
        #pragma unroll 16
        for (int jj = 0; jj < 64; ++jj) {
            const int j = q * 64 + jj;
            X[cc][j][c16] = valid ? rp[j] : 0.0f;
        }
    }
    __syncthreads();

    v8f acc[2][CT];
    #pragma unroll
    for (int rr = 0; rr < 2; ++rr)
        #pragma unroll
        for (int cc = 0; cc < CT; ++cc)
            acc[rr][cc] = (v8f){0.f,0.f,0.f,0.f,0.f,0.f,0.f,0.f};

    #pragma unroll 4
    for (int k0 = 0; k0 < D; k0 += 4) {
        const int kb = k0 + lhalf * 2;
        v2f a0, a1;
        a0[0] = arow0[kb]; a0[1] = arow0[kb + 1];
        a1[0] = arow1[kb]; a1[1] = arow1[kb + 1];
        #pragma unroll
        for (int cc = 0; cc < CT; ++cc) {
            v2f bb;
            bb[0] = X[cc][kb][l15];
            bb[1] = X[cc][kb + 1][l15];
            acc[0][cc] = __builtin_amdgcn_wmma_f32_16x16x4_f32(
                false, a0, false, bb, (short)0, acc[0][cc], false, false);
            acc[1][cc] = __builtin_amdgcn_wmma_f32_16x16x4_f32(
                false, a1, false, bb, (short)0, acc[1][cc], false, false);
        }
    }

    #pragma unroll
    for (int cc = 0; cc < CT; ++cc) {
        const int c = cc * 16 + l15;
        float s = 0.0f;
        if (vcol[c]) {
            const float* ub = U + (size_t)bcol[c] * D;
            #pragma unroll
            for (int rr = 0; rr < 2; ++rr) {
                const int rbase = (wave + rr * 4) * 16 + lhalf * 8;
                #pragma unroll
                for (int g = 0; g < 8; ++g) {
                    const float dv = acc[rr][cc][g] + ub[rbase + g];
                    s += dv * dv;
                }
            }
        }
        psum[wave * 2 + lhalf][c] = s;
    }
    __syncthreads();

    if (tid < COLS && vcol[tid]) {
        float tot = 0.0f;
        #pragma unroll
        for (int i = 0; i < 8; ++i) tot += psum[i][tid];   // fixed order: deterministic
        out[bcol[tid]] = sqrtf(tot);
    }
}

// ---------------- host launcher ----------------

extern "C" void kernel_launch(void* const* d_in, const int* in_sizes, int n_in,
                              void* d_out, int out_size, void* d_ws, size_t ws_size,
                              hipStream_t stream) {
    const int*   heads = (const int*)d_in[0];
    const int*   rels  = (const int*)d_in[1];
    const int*   tails = (const int*)d_in[2];
    const int*   etype = (const int*)d_in[3];
    const int*   rtype = (const int*)d_in[4];
    const float* eemb  = (const float*)d_in[5];
    const float* remb  = (const float*)d_in[6];
    const float* eproj = (const float*)d_in[7];
    const float* rproj = (const float*)d_in[8];
    float*       out   = (float*)d_out;

    // workspace carve-up: U (B*D f32) + perms + counters
    float* U     = (float*)d_ws;
    int*   permE = (int*)(U + (size_t)BATCH * D);
    int*   permR = permE + BATCH;
    int*   cntE  = permR + BATCH;
    int*   offE  = cntE + NET;
    int*   curE  = offE + NET;
    int*   cntR  = curE + NET;
    int*   offR  = cntR + NRT;
    int*   curR  = offR + NRT;

    k_init<<<1, 128, 0, stream>>>(cntE, cntR);
    k_count<<<BATCH / 256, 256, 0, stream>>>(etype, rtype, cntE, cntR);
    k_scan<<<1, 64, 0, stream>>>(cntE, offE, curE, cntR, offR, curR);
    k_scatter<<<BATCH / 256, 256, 0, stream>>>(etype, rtype, curE, curR, permE, permR);

    k_gemm_e<<<dim3(NET, MAXT), 128, 0, stream>>>(heads, tails, eemb, eproj,
                                                  cntE, offE, permE, U);
    k_gemm_r<<<dim3(NRT, MAXT), 128, 0, stream>>>(rels, remb, rproj,
                                                  cntR, offR, permR, U, out);
}